// HypAgg_40415642255634
// MI455X (gfx1250) — compile-verified
//
#include <hip/hip_runtime.h>
#include <hip/hip_bf16.h>
#include <math.h>

// HGCN HypAgg on MI455X (gfx1250), fused flash-attention style.
// support = G_i * x_i + (att .* h) @ x  with all pair scalars derived from
// the Gram matrix X X^T. Both GEMMs use V_WMMA_F32_16X16X4_F32 (full f32).

typedef __attribute__((ext_vector_type(2))) float v2f;
typedef __attribute__((ext_vector_type(8))) float v8f;

#define N_NODES 1024
#define DIM     128
#define MIN_NORM 1e-15f
#define PROJ_EPS 4e-3f

__device__ __forceinline__ float artanh_c(float z) {
    // z >= 0 here; clip like the reference
    z = fminf(z, 1.f - 1e-7f);
    return 0.5f * __logf((1.f + z) / (1.f - z));
}

// ---------------- kernel 1: per-row stats ----------------
// x2[r] = |x_r|^2 ; left/right = <logmap0(x_r), w_lo/hi>
__global__ void row_stats_kernel(const float* __restrict__ x,
                                 const float* __restrict__ att_w,
                                 float* __restrict__ x2s,
                                 float* __restrict__ lefts,
                                 float* __restrict__ rights) {
    __shared__ float sm0[DIM], sm1[DIM], sm2[DIM];
    const int r = blockIdx.x, t = threadIdx.x;
    float v = x[r * DIM + t];
    sm0[t] = v * v;
    sm1[t] = v * att_w[t];
    sm2[t] = v * att_w[DIM + t];
    __syncthreads();
    for (int s = 64; s > 0; s >>= 1) {
        if (t < s) { sm0[t] += sm0[t + s]; sm1[t] += sm1[t + s]; sm2[t] += sm2[t + s]; }
        __syncthreads();
    }
    if (t == 0) {
        float x2 = sm0[0];
        float pn = fmaxf(sqrtf(x2), MIN_NORM);
        float sc = artanh_c(pn) / pn;                 // logmap0 radial scale
        x2s[r] = x2;
        lefts[r] = sc * sm1[0];
        rights[r] = sc * sm2[0];
    }
}

// ---------------- kernel 2: fused attention-style aggregation ----------------
// block = 256 threads (8 waves). One 16-row i-tile per block; waves split j-tiles.
__global__ __launch_bounds__(256, 1)
void hyp_agg_kernel(const float* __restrict__ x, const float* __restrict__ adj,
                    const float* __restrict__ att_b_p,
                    const float* __restrict__ x2s, const float* __restrict__ lefts,
                    const float* __restrict__ rights,
                    float* __restrict__ out) {
    __shared__ float lds_supp[16 * DIM];   // 8 KB  cross-wave support accumulator
    __shared__ float lds_G[16];            //        cross-wave diag-coef accumulator
    __shared__ float lds_H[8][16 * 16];    // 8 KB  per-wave H tile (C-layout -> A-layout relay)

    const int tid  = threadIdx.x;
    const int wave = tid >> 5;
    const int lane = tid & 31;
    const int hi   = lane >> 4;   // 0 or 1 : lane half
    const int ln   = lane & 15;
    const int i0   = blockIdx.x * 16;
    const float att_b = att_b_p[0];

    for (int k = tid; k < 16 * DIM; k += 256) lds_supp[k] = 0.f;
    if (tid < 16) lds_G[tid] = 0.f;
    __syncthreads();

    // loop-invariant per-row (i) scalars; acc row m = v + 8*hi (WMMA C/D layout)
    float ui[8], li[8];
#pragma unroll
    for (int v = 0; v < 8; ++v) {
        int row = i0 + v + 8 * hi;
        ui[v] = x2s[row];
        li[v] = lefts[row];
    }

    float gacc[8];
#pragma unroll
    for (int v = 0; v < 8; ++v) gacc[v] = 0.f;
    v8f sacc[8];
#pragma unroll
    for (int dt = 0; dt < 8; ++dt) sacc[dt] = {};

    for (int jt = wave; jt < N_NODES / 16; jt += 8) {
        const int j0 = jt * 16;

        // ---- Gram tile: dot[16i x 16j] = X_i X_j^T via 32 f32 WMMAs (K=128) ----
        v8f dacc = {};
#pragma unroll
        for (int kf = 0; kf < 32; ++kf) {
            const int col = 4 * kf + 2 * hi;
            v2f a = *(const v2f*)(x + (i0 + ln) * DIM + col);  // A[m][k] = X[i0+m][k]
            v2f b = *(const v2f*)(x + (j0 + ln) * DIM + col);  // B[k][n] = X[j0+n][k]
            dacc = __builtin_amdgcn_wmma_f32_16x16x4_f32(false, a, false, b,
                                                         (short)0, dacc, false, false);
        }

        // ---- pointwise hyperbolic scalar field on the tile ----
        const int j = j0 + ln;
        const float vj = x2s[j];
        const float rj = rights[j];
#pragma unroll
        for (int v = 0; v < 8; ++v) {
            const int m = v + 8 * hi;
            float dot = dacc[v];
            float Af = 1.f - 2.f * dot + vj;       // mobius_add(-x_i, x_j) pieces, c=1
            float Bf = 1.f - ui[v];
            float den = fmaxf(1.f - 2.f * dot + ui[v] * vj, MIN_NORM);
            float sn2 = fmaxf(Af * Af * ui[v] - 2.f * Af * Bf * dot + Bf * Bf * vj, 0.f)
                        / (den * den);
            float sn = fmaxf(sqrtf(sn2), MIN_NORM);
            float coef = fmaxf(Bf, MIN_NORM);      // 2/(sqrt_c*lambda_i) = clip(1-u)
            float s = coef * artanh_c(sn) / sn;
            float att = adj[(i0 + m) * N_NODES + j]
                        / (1.f + __expf(-(li[v] + rj + att_b)));
            gacc[v] += att * s * (-Af / den);                  // coef on x_i
            lds_H[wave][m * 16 + ln] = att * s * (Bf / den);   // coef on x_j
        }

        // ---- aggregation: support[16 x 128] += H(16x16) * X[j0:j0+16, :] ----
#pragma unroll
        for (int kf = 0; kf < 4; ++kf) {
            const int k0 = 4 * kf;
            v2f ha = *(const v2f*)&lds_H[wave][ln * 16 + k0 + 2 * hi]; // A[m][k]=H[m][k]
            const int jr = j0 + k0 + 2 * hi;
#pragma unroll
            for (int dt = 0; dt < 8; ++dt) {
                const int d = dt * 16 + ln;
                v2f b;                                  // B[k][n] = X[j0+k][d0+n]
                b.x = x[jr * DIM + d];
                b.y = x[(jr + 1) * DIM + d];
                sacc[dt] = __builtin_amdgcn_wmma_f32_16x16x4_f32(false, ha, false, b,
                                                                 (short)0, sacc[dt],
                                                                 false, false);
            }
        }
    }

    // ---- cross-wave reduction via LDS float atomics (ds_add_f32) ----
#pragma unroll
    for (int v = 0; v < 8; ++v)
        atomicAdd(&lds_G[v + 8 * hi], gacc[v]);
#pragma unroll
    for (int dt = 0; dt < 8; ++dt)
#pragma unroll
        for (int v = 0; v < 8; ++v)
            atomicAdd(&lds_supp[(v + 8 * hi) * DIM + dt * 16 + ln], sacc[dt][v]);
    __syncthreads();

    // ---- epilogue: expmap_{x_i}(support_i) + proj ; one thread per row ----
    if (tid < 16) {
        const int i = i0 + tid;
        const float u = x2s[i];
        const float G = lds_G[tid];
        float un2 = 0.f, xs = 0.f;
        for (int d = 0; d < DIM; ++d) {
            float sv = G * x[i * DIM + d] + lds_supp[tid * DIM + d];
            lds_supp[tid * DIM + d] = sv;
            un2 += sv * sv;
            xs  += x[i * DIM + d] * sv;
        }
        float un  = fmaxf(sqrtf(un2), MIN_NORM);
        float lam = 2.f / fmaxf(1.f - u, MIN_NORM);
        float kf  = tanhf(0.5f * lam * un) / un;   // second = kf * support
        float y2  = kf * kf * un2;
        float xy  = kf * xs;
        float numA = 1.f + 2.f * xy + y2;          // mobius_add(x, second), c=1
        float numB = 1.f - u;
        float den  = fmaxf(1.f + 2.f * xy + u * y2, MIN_NORM);
        float res2 = (numA * numA * u + 2.f * numA * numB * xy + numB * numB * y2)
                     / (den * den);
        float nrm = fmaxf(sqrtf(res2), MIN_NORM);
        float maxnorm = 1.f - PROJ_EPS;
        float scale = (nrm > maxnorm) ? (maxnorm / nrm) : 1.f;
        for (int d = 0; d < DIM; ++d)
            out[i * DIM + d] =
                scale * (numA * x[i * DIM + d] + numB * kf * lds_supp[tid * DIM + d]) / den;
    }
}

extern "C" void kernel_launch(void* const* d_in, const int* in_sizes, int n_in,
                              void* d_out, int out_size, void* d_ws, size_t ws_size,
                              hipStream_t stream) {
    const float* x     = (const float*)d_in[0];   // [N, D]
    const float* adj   = (const float*)d_in[1];   // [N, N]
    const float* att_w = (const float*)d_in[2];   // [2D]
    const float* att_b = (const float*)d_in[3];   // [1]
    float* out = (float*)d_out;
    float* ws  = (float*)d_ws;                    // 3*N floats = 12 KB
    float* x2s    = ws;
    float* lefts  = ws + N_NODES;
    float* rights = ws + 2 * N_NODES;

    row_stats_kernel<<<N_NODES, DIM, 0, stream>>>(x, att_w, x2s, lefts, rights);
    hyp_agg_kernel<<<N_NODES / 16, 256, 0, stream>>>(x, adj, att_b,
                                                     x2s, lefts, rights, out);
}